// GCMultiHeadAttention_53704271069644
// MI455X (gfx1250) — compile-verified
//
#include <hip/hip_runtime.h>
#include <hip/hip_bf16.h>

// ---------------------------------------------------------------------------
// GC multi-head attention for MI455X (gfx1250), fp32 via V_WMMA_F32_16X16X4_F32
//
// Shapes: H=8 heads, B=8, N=1024, D=128 input dim, KD=16 key/val dim.
// Workspace (float): 8 projections (H,B*N,16) + heads_node + heads_c = 40 MB.
// ---------------------------------------------------------------------------

#define HN 8
#define BN 8
#define NN 1024
#define DN 128
#define KD 16
#define ROWS (BN * NN)   // 8192

typedef __attribute__((ext_vector_type(2))) float v2f;
typedef __attribute__((ext_vector_type(8))) float v8f;

// D = A(16x4) * B(4x16) + C, fp32.  8-arg form:
// (neg_a, A, neg_b, B, c_mod, C, reuse_a, reuse_b)
__device__ __forceinline__ v8f wmma_f32_k4(v2f a, v2f b, v8f c) {
  return __builtin_amdgcn_wmma_f32_16x16x4_f32(false, a, false, b, (short)0, c,
                                               false, false);
}

// ---------------------------------------------------------------------------
// Projection: Out[h, row, k] = sum_d X[row, d] * W[h, d, k]
// X: (8192,128)  W: (8,128,16)  Out: (8, 8192, 16)
// One block = 64-row tile of X staged in LDS; wave w computes head w,
// 4 sub-tiles of 16x16, K=128 contraction = 32 chained 16x16x4 WMMAs each.
// ---------------------------------------------------------------------------
__global__ __launch_bounds__(256) void proj_kernel(const float* __restrict__ X,
                                                   const float* __restrict__ W,
                                                   float* __restrict__ Out) {
  __shared__ float xs[64 * DN];  // 32 KB
  const int tid  = threadIdx.x;
  const int row0 = blockIdx.x * 64;

  const float4* Xv  = (const float4*)(X + (size_t)row0 * DN);
  float4*       xsv = (float4*)xs;
#pragma unroll
  for (int i = 0; i < 8; ++i) xsv[tid + i * 256] = Xv[tid + i * 256];
  __syncthreads();

  const int w = tid >> 5;  // wave id == head
  const int lane = tid & 31, lo = lane & 15, hi = lane >> 4;
  const float* Wh = W + (size_t)w * (DN * KD);

  v8f acc0 = {0,0,0,0,0,0,0,0}, acc1 = acc0, acc2 = acc0, acc3 = acc0;
  v8f* acc[4] = {&acc0, &acc1, &acc2, &acc3};

  for (int kc = 0; kc < DN; kc += 4) {
    v2f b;
    b.x = Wh[(kc + 2 * hi) * KD + lo];       // B[K=kc+2hi][N=lo]
    b.y = Wh[(kc + 1 + 2 * hi) * KD + lo];   // B[K=kc+1+2hi][N=lo]
#pragma unroll
    for (int ms = 0; ms < 4; ++ms) {
      const float* ap = &xs[(ms * 16 + lo) * DN + kc + 2 * hi];
      v2f a; a.x = ap[0]; a.y = ap[1];       // A[M=lo][K=kc+2hi, +1]
      *acc[ms] = wmma_f32_k4(a, b, *acc[ms]);
    }
  }
#pragma unroll
  for (int ms = 0; ms < 4; ++ms)
#pragma unroll
    for (int v = 0; v < 8; ++v)
      Out[(size_t)w * (ROWS * KD) + (size_t)(row0 + ms * 16 + hi * 8 + v) * KD + lo] =
          (*acc[ms])[v];
}

// ---------------------------------------------------------------------------
// Flash attention, fp32.  Q,K,V: (H*B, N, 16) flat.  One block per
// (head*batch, 128-query tile); each of 8 waves owns 16 query rows.
// K/V streamed through LDS in 256-key chunks.  Row stats reduced with
// 16-lane shuffles (C/D fragment rows live across a 16-lane half-group).
// ---------------------------------------------------------------------------
__global__ __launch_bounds__(256) void attn_kernel(
    const float* __restrict__ Qp, const float* __restrict__ Kp,
    const float* __restrict__ Vp, const unsigned char* __restrict__ mask,
    float* __restrict__ Out, int accumulate) {
  __shared__ float ks[256 * KD];       // 16 KB
  __shared__ float vs[256 * KD];       // 16 KB
  __shared__ float pbuf[8][16 * 16];   // 8 KB, per-wave P-tile transpose bounce

  const int hb = blockIdx.x;           // h*B + b
  const int b  = hb & (BN - 1);
  const int tid = threadIdx.x;
  const int w = tid >> 5, lane = tid & 31, lo = lane & 15, hi = lane >> 4;
  const int q0 = blockIdx.y * 128 + w * 16;
  const size_t base = (size_t)hb * (NN * KD);
  const float norm = 0.25f;            // 1/sqrt(16)

  // Persistent Q A-fragments: 4 K-chunks of the 16-wide contraction dim.
  v2f qa[4];
#pragma unroll
  for (int c = 0; c < 4; ++c) {
    const float* qp = Qp + base + (size_t)(q0 + lo) * KD + 4 * c + 2 * hi;
    qa[c].x = qp[0]; qa[c].y = qp[1];
  }

  v8f o = {0,0,0,0,0,0,0,0};
  float mrow[8], lrow[8];
#pragma unroll
  for (int v = 0; v < 8; ++v) { mrow[v] = -3.0e38f; lrow[v] = 0.f; }

  for (int chunk = 0; chunk < 4; ++chunk) {
    __syncthreads();
    {
      const float4* Ksrc = (const float4*)(Kp + base + (size_t)chunk * 256 * KD);
      const float4* Vsrc = (const float4*)(Vp + base + (size_t)chunk * 256 * KD);
      float4* kd = (float4*)ks; float4* vd = (float4*)vs;
#pragma unroll
      for (int i = 0; i < 4; ++i) {
        kd[tid + i * 256] = Ksrc[tid + i * 256];
        vd[tid + i * 256] = Vsrc[tid + i * 256];
      }
      if (chunk < 3) {  // gfx1250 global_prefetch of the next K/V chunk
        __builtin_prefetch(Kp + base + (size_t)(chunk + 1) * 256 * KD + tid * 16, 0, 0);
        __builtin_prefetch(Vp + base + (size_t)(chunk + 1) * 256 * KD + tid * 16, 0, 0);
      }
    }
    __syncthreads();

    for (int j = 0; j < 16; ++j) {          // 16-key blocks within chunk
      const int kb   = j * 16;              // key base in LDS
      const int keyg = chunk * 256 + kb;    // global key base

      // S = Q * K^T  (16x16 tile, K-dim 16 => 4 chained WMMAs)
      v8f s = {0,0,0,0,0,0,0,0};
#pragma unroll
      for (int c = 0; c < 4; ++c) {
        const float* kp = &ks[(kb + lo) * KD + 4 * c + 2 * hi];
        v2f kf; kf.x = kp[0]; kf.y = kp[1];  // B[d][key]=K[key][d]
        s = wmma_f32_k4(qa[c], kf, s);
      }

      // scale + mask (element (M=v+8hi, N=lo) => query q0+v+8hi, key keyg+lo)
      float sv[8];
#pragma unroll
      for (int v = 0; v < 8; ++v) {
        float x = s[v] * norm;
        if (mask) {
          const int qr  = q0 + v + 8 * hi;
          const int key = keyg + lo;
          if (mask[(size_t)(b * NN + qr) * NN + key]) x = -3.0e38f;
        }
        sv[v] = x;
      }

      // online softmax: row max / row sum over the 16-lane half-group
#pragma unroll
      for (int v = 0; v < 8; ++v) {
        float mx = sv[v];
#pragma unroll
        for (int sh = 1; sh < 16; sh <<= 1)
          mx = fmaxf(mx, __shfl_xor(mx, sh, 16));
        const float mn    = fmaxf(mrow[v], mx);
        const float alpha = __expf(mrow[v] - mn);
        const float pv    = (sv[v] < -1.0e37f) ? 0.f : __expf(sv[v] - mn);
        float rs = pv;
#pragma unroll
        for (int sh = 1; sh < 16; sh <<= 1)
          rs += __shfl_xor(rs, sh, 16);
        lrow[v] = lrow[v] * alpha + rs;
        o[v]    = o[v] * alpha;
        mrow[v] = mn;
        pbuf[w][(v + 8 * hi) * 16 + lo] = pv;  // D-layout -> row-major 16x16
      }

      // O += P * V  (P re-read from LDS in A-layout)
#pragma unroll
      for (int c = 0; c < 4; ++c) {
        const int kc = 4 * c;
        const float* pp = &pbuf[w][lo * 16 + kc + 2 * hi];
        v2f pa; pa.x = pp[0]; pa.y = pp[1];            // A[M=lo][K=kc+2hi,+1]
        const float* vp = &vs[(kb + kc + 2 * hi) * KD + lo];
        v2f vb; vb.x = vp[0]; vb.y = vp[KD];           // B[K][N=lo], rows K,K+1
        o = wmma_f32_k4(pa, vb, o);
      }
    }
  }

  // epilogue: normalize, guard fully-masked rows, write heads (h,b,n,16)
#pragma unroll
  for (int v = 0; v < 8; ++v) {
    const int qr  = q0 + v + 8 * hi;
    const float val = (lrow[v] > 0.f) ? o[v] / lrow[v] : 0.f;
    const size_t idx = base + (size_t)qr * KD + lo;
    if (accumulate) Out[idx] += val; else Out[idx] = val;
  }
}

// ---------------------------------------------------------------------------
// Output projection: Out[row, e] = sum_h sum_v Heads[h,row,v] * Wout[h,v,e]
// == (8192 x 128) * (128 x 128) with heads concatenated over (h,v).
// ---------------------------------------------------------------------------
__global__ __launch_bounds__(256) void outproj_kernel(
    const float* __restrict__ Heads, const float* __restrict__ Wout,
    float* __restrict__ Out) {
  __shared__ float xs[64 * DN];  // heads_cat 64x128 tile, 32 KB
  const int tid  = threadIdx.x;
  const int row0 = blockIdx.x * 64;

  // stage heads_cat[row][h*16+v] = Heads[h, row0+row, v]
#pragma unroll
  for (int i = 0; i < 8; ++i) {
    const int f   = tid + i * 256;  // float4 index (2048 total)
    const int row = f >> 5;         // 32 float4 per 128-col row
    const int c4  = f & 31;
    const int h   = c4 >> 2;
    const int v4  = c4 & 3;
    const float4* src = (const float4*)(Heads + (size_t)h * (ROWS * KD) +
                                        (size_t)(row0 + row) * KD + v4 * 4);
    ((float4*)xs)[f] = *src;
  }
  __syncthreads();

  const int w = tid >> 5, lane = tid & 31, lo = lane & 15, hi = lane >> 4;
  const int n0 = w * 16;  // output column tile

  v8f acc0 = {0,0,0,0,0,0,0,0}, acc1 = acc0, acc2 = acc0, acc3 = acc0;
  v8f* acc[4] = {&acc0, &acc1, &acc2, &acc3};

  for (int kc = 0; kc < DN; kc += 4) {
    const int h = kc >> 4;
    const int v = kc & 15;  // 0,4,8,12 => v+2hi+1 stays within this head
    const float* wp = Wout + (size_t)h * (KD * DN) + (size_t)(v + 2 * hi) * DN + n0 + lo;
    v2f b; b.x = wp[0]; b.y = wp[DN];
#pragma unroll
    for (int ms = 0; ms < 4; ++ms) {
      const float* ap = &xs[(ms * 16 + lo) * DN + kc + 2 * hi];
      v2f a; a.x = ap[0]; a.y = ap[1];
      *acc[ms] = wmma_f32_k4(a, b, *acc[ms]);
    }
  }
#pragma unroll
  for (int ms = 0; ms < 4; ++ms)
#pragma unroll
    for (int v = 0; v < 8; ++v)
      Out[(size_t)(row0 + ms * 16 + v + 8 * hi) * DN + n0 + lo] = (*acc[ms])[v];
}

// ---------------------------------------------------------------------------
extern "C" void kernel_launch(void* const* d_in, const int* in_sizes, int n_in,
                              void* d_out, int out_size, void* d_ws, size_t ws_size,
                              hipStream_t stream) {
  (void)in_sizes; (void)n_in; (void)out_size; (void)ws_size;

  const float* q_n  = (const float*)d_in[0];
  const float* q_c  = (const float*)d_in[1];
  const unsigned char* mask = (const unsigned char*)d_in[2];  // bool (B,N,N)
  const float* W_query_n = (const float*)d_in[3];
  const float* W_key_nn  = (const float*)d_in[4];
  const float* W_val_nn  = (const float*)d_in[5];
  const float* W_key_c   = (const float*)d_in[6];
  const float* W_val_c   = (const float*)d_in[7];
  const float* W_query_c = (const float*)d_in[8];
  const float* W_key_n   = (const float*)d_in[9];
  const float* W_val_n   = (const float*)d_in[10];
  const float* W_out_node  = (const float*)d_in[11];
  const float* W_out_color = (const float*)d_in[12];

  float* ws = (float*)d_ws;
  const size_t P = (size_t)HN * ROWS * KD;  // 1,048,576 floats per projection
  float* Q_n  = ws + 0 * P;  float* K_nn = ws + 1 * P;  float* V_nn = ws + 2 * P;
  float* K_nc = ws + 3 * P;  float* V_nc = ws + 4 * P;  float* Q_c  = ws + 5 * P;
  float* K_n  = ws + 6 * P;  float* V_n  = ws + 7 * P;
  float* heads_node = ws + 8 * P;
  float* heads_c    = ws + 9 * P;   // total 40 MB of d_ws

  const dim3 blk(256);

  proj_kernel<<<128, blk, 0, stream>>>(q_n, W_query_n, Q_n);
  proj_kernel<<<128, blk, 0, stream>>>(q_n, W_key_nn,  K_nn);
  proj_kernel<<<128, blk, 0, stream>>>(q_n, W_val_nn,  V_nn);
  proj_kernel<<<128, blk, 0, stream>>>(q_c, W_key_c,   K_nc);
  proj_kernel<<<128, blk, 0, stream>>>(q_c, W_val_c,   V_nc);
  proj_kernel<<<128, blk, 0, stream>>>(q_c, W_query_c, Q_c);
  proj_kernel<<<128, blk, 0, stream>>>(q_n, W_key_n,   K_n);
  proj_kernel<<<128, blk, 0, stream>>>(q_n, W_val_n,   V_n);

  const dim3 agrid(HN * BN, NN / 128);  // (64, 8)
  attn_kernel<<<agrid, blk, 0, stream>>>(Q_n, K_nn, V_nn, mask,    heads_node, 0);
  attn_kernel<<<agrid, blk, 0, stream>>>(Q_n, K_nc, V_nc, nullptr, heads_node, 1);
  attn_kernel<<<agrid, blk, 0, stream>>>(Q_c, K_n,  V_n,  nullptr, heads_c,    0);

  float* out_node  = (float*)d_out;                          // (B,N,128)
  float* out_color = out_node + (size_t)BN * NN * DN;        // (B,N,128)
  outproj_kernel<<<128, blk, 0, stream>>>(heads_node, W_out_node,  out_node);
  outproj_kernel<<<128, blk, 0, stream>>>(heads_c,    W_out_color, out_color);
}